// Preprocess_47931835023561
// MI455X (gfx1250) — compile-verified
//
#include <hip/hip_runtime.h>
#include <stdint.h>

// Preprocess pipeline: rescale + rotate -> zoom -> shift + hflip
// B=64, H=W=224, C=3, f32. Memory-bound (~230MB traffic, ~1.3 GFLOP): no
// matmul structure -> WMMA not applicable. Design targets the MI455X memory
// system instead: 3 fused passes ping-ponging through buffers that fit in the
// 192MB L2, with the axis-aligned passes staging their two source rows into
// LDS via gfx1250 async-to-LDS loads (ASYNCcnt path), and a non-temporal
// store policy on the final (never re-read) output.

namespace {

constexpr int   B      = 64;
constexpr int   H      = 224;
constexpr int   W      = 224;
constexpr int   C      = 3;
constexpr float FACTOR = 0.2f;
constexpr float SCALE  = 1.0f / 255.0f;
constexpr float TWO_PI = 6.283185307179586f;

// TF 'reflect' fill mode: m = mod(i, 2n); m >= n ? 2n-1-m : m  (floor-mod)
__device__ __forceinline__ int reflect_idx(int i, int n) {
  const int p = 2 * n;
  int m = i % p;
  m += (m >> 31) & p;            // branchless fix-up for negative remainders
  return (m >= n) ? (p - 1 - m) : m;
}

// ---------------------------------------------------------------------------
// Pass 1: rotation warp with the 1/255 rescale folded into the taps.
// One thread = one output pixel (3 channels). 2D-varying source coords ->
// plain global gathers; neighboring lanes sample neighboring pixels, so the
// 4 taps are L2/WGP$-local (whole 38.5MB batch fits in L2 ~5x over).
// ---------------------------------------------------------------------------
__global__ __launch_bounds__(256) void rot_rescale_kernel(
    const float* __restrict__ img, const float* __restrict__ angle_u,
    float* __restrict__ dst) {
  const int b   = blockIdx.y;
  const int pix = blockIdx.x * 256 + threadIdx.x;   // 196 * 256 == H*W
  const int y   = pix / W;
  const int x   = pix - y * W;

  const float cx = (W - 1) * 0.5f;
  const float cy = (H - 1) * 0.5f;
  const float theta = (2.0f * angle_u[b] - 1.0f) * (FACTOR * TWO_PI);
  float s, c;
  sincosf(theta, &s, &c);        // accurate version; once per thread

  const float rx = (float)x - cx;
  const float ry = (float)y - cy;
  const float xs = c * rx - s * ry + cx;
  const float ys = s * rx + c * ry + cy;

  const float x0f = floorf(xs), y0f = floorf(ys);
  const float fx = xs - x0f,    fy = ys - y0f;
  const int   x0 = (int)x0f,    y0 = (int)y0f;
  const int x0r = reflect_idx(x0, W),     x1r = reflect_idx(x0 + 1, W);
  const int y0r = reflect_idx(y0, H),     y1r = reflect_idx(y0 + 1, H);

  const float* base = img + (size_t)b * (H * W * C);
  const float* pa = base + (y0r * W + x0r) * C;
  const float* pb = base + (y0r * W + x1r) * C;
  const float* pc = base + (y1r * W + x0r) * C;
  const float* pd = base + (y1r * W + x1r) * C;
  float* o = dst + (size_t)b * (H * W * C) + (size_t)pix * C;

  const float omfx = 1.0f - fx;
  const float omfy = 1.0f - fy;
#pragma unroll
  for (int ch = 0; ch < C; ++ch) {
    const float Ia = pa[ch] * SCALE, Ib = pb[ch] * SCALE;
    const float Ic = pc[ch] * SCALE, Id = pd[ch] * SCALE;
    const float top = Ia * omfx + Ib * fx;
    const float bot = Ic * omfx + Id * fx;
    o[ch] = top * omfy + bot * fy;
  }
}

// ---------------------------------------------------------------------------
// Axis-aligned warp of one output row. The row's source y is uniform across
// the block, so the two (y-reflected) source rows are staged into LDS with
// gfx1250 async-to-LDS b128 loads (16B chunks; row = 2688B = 168 chunks,
// 16B-aligned in both global and LDS), then all x taps read from LDS.
// NT_STORE=true issues non-temporal stores (final pass: output never re-read,
// keep the L2 free for the intermediate buffer still being streamed).
// ---------------------------------------------------------------------------
template <bool NT_STORE>
__device__ __forceinline__ void axis_warp_row(
    const float* __restrict__ src, float* __restrict__ dst,
    int b, int y, float xscale, float xoff, float ys) {
  constexpr int ROWBYTES = W * C * 4;          // 2688
  constexpr int CHUNKS   = ROWBYTES / 16;      // 168 per row
  __shared__ __align__(16) float rows[2][W * C];

  const float y0f = floorf(ys);
  const float fy  = ys - y0f;
  const int   y0  = (int)y0f;
  const int y0r = reflect_idx(y0, H);
  const int y1r = reflect_idx(y0 + 1, H);
  const float* r0 = src + ((size_t)b * H + y0r) * (W * C);
  const float* r1 = src + ((size_t)b * H + y1r) * (W * C);

  // Async stage: each lane copies one 16B chunk memory->LDS (ASYNCcnt path).
  for (int kc = threadIdx.x; kc < 2 * CHUNKS; kc += blockDim.x) {
    const int rsel    = (kc >= CHUNKS);
    const int byteoff = (kc - rsel * CHUNKS) * 16;
    const char* g = (const char*)(rsel ? r1 : r0) + byteoff;
    const unsigned lds =
        (unsigned)(uintptr_t)((char*)&rows[rsel][0] + byteoff);
    asm volatile("global_load_async_to_lds_b128 %0, %1, off"
                 :: "v"(lds), "v"(g) : "memory");
  }
  asm volatile("s_wait_asynccnt 0" ::: "memory");
  __syncthreads();

  const float omfy = 1.0f - fy;
  for (int x = threadIdx.x; x < W; x += blockDim.x) {
    const float xs  = xscale * (float)x + xoff;
    const float x0f = floorf(xs);
    const float fx  = xs - x0f;
    const int   x0  = (int)x0f;
    const int x0r = reflect_idx(x0, W) * C;
    const int x1r = reflect_idx(x0 + 1, W) * C;
    const float omfx = 1.0f - fx;
    float* o = dst + (((size_t)b * H + y) * W + x) * C;
#pragma unroll
    for (int ch = 0; ch < C; ++ch) {
      const float Ia = rows[0][x0r + ch], Ib = rows[0][x1r + ch];
      const float Ic = rows[1][x0r + ch], Id = rows[1][x1r + ch];
      const float top = Ia * omfx + Ib * fx;
      const float bot = Ic * omfx + Id * fx;
      const float v   = top * omfy + bot * fy;
      if constexpr (NT_STORE) {
        __builtin_nontemporal_store(v, o + ch);
      } else {
        o[ch] = v;
      }
    }
  }
}

// Pass 2: zoom. xs = zw*(x-cx)+cx, ys = zh*(y-cy)+cy
__global__ __launch_bounds__(256) void zoom_kernel(
    const float* __restrict__ src, const float* __restrict__ zoom_u,
    float* __restrict__ dst) {
  const int b = blockIdx.y;
  const int y = blockIdx.x;
  const float cx = (W - 1) * 0.5f;
  const float cy = (H - 1) * 0.5f;
  const float zh = 1.0f + (2.0f * zoom_u[2 * b + 0] - 1.0f) * FACTOR;
  const float zw = 1.0f + (2.0f * zoom_u[2 * b + 1] - 1.0f) * FACTOR;
  const float ys = zh * ((float)y - cy) + cy;
  axis_warp_row<false>(src, dst, b, y, zw, cx - zw * cx, ys);
}

// Pass 3: shift with hflip folded into the sample coordinate:
// final[y,x] = bilinear(I2, (flip ? W-1-x : x) - dx, y - dy)
__global__ __launch_bounds__(256) void shift_flip_kernel(
    const float* __restrict__ src, const float* __restrict__ shift_u,
    const float* __restrict__ flip_u, float* __restrict__ dst) {
  const int b = blockIdx.y;
  const int y = blockIdx.x;
  const float dy = (2.0f * shift_u[2 * b + 0] - 1.0f) * FACTOR * (float)H;
  const float dx = (2.0f * shift_u[2 * b + 1] - 1.0f) * FACTOR * (float)W;
  const bool  flip   = flip_u[b] > 0.5f;
  const float xscale = flip ? -1.0f : 1.0f;
  const float xoff   = flip ? ((float)(W - 1) - dx) : (-dx);
  axis_warp_row<true>(src, dst, b, y, xscale, xoff, (float)y - dy);
}

}  // namespace

extern "C" void kernel_launch(void* const* d_in, const int* in_sizes, int n_in,
                              void* d_out, int out_size, void* d_ws, size_t ws_size,
                              hipStream_t stream) {
  (void)in_sizes; (void)n_in; (void)out_size; (void)ws_size;
  const float* img     = (const float*)d_in[0];
  const float* angle_u = (const float*)d_in[1];
  const float* zoom_u  = (const float*)d_in[2];
  const float* shift_u = (const float*)d_in[3];
  const float* flip_u  = (const float*)d_in[4];
  float* out = (float*)d_out;
  float* tmp = (float*)d_ws;   // needs B*H*W*C*4 = ~38.5 MB of scratch

  // Ping-pong: img -> out -> ws -> out. Both 38.5MB buffers fit in the
  // 192MB L2, so passes 2 and 3 read mostly from on-die cache.
  rot_rescale_kernel<<<dim3((H * W) / 256, B), 256, 0, stream>>>(img, angle_u, out);
  zoom_kernel      <<<dim3(H, B),             256, 0, stream>>>(out, zoom_u, tmp);
  shift_flip_kernel<<<dim3(H, B),             256, 0, stream>>>(tmp, shift_u, flip_u, out);
}